// RetinaNet_55276229099833
// MI455X (gfx1250) — compile-verified
//
#include <hip/hip_runtime.h>
#include <math.h>

#define NUM_CLS 80
#define ANC 9
#define NLVL 5
#define NIMG 2
#define TOPK 1000
#define MAXDET 100
#define NCAND 5000            // NLVL * TOPK per image
#define NBINS 4096
#define IMGSZ 1024.0f
#define NMS_IOU 0.5f
#define LOGIT_TH (-2.9444389791664403f)   // log(0.05/0.95)
#define SCALE_CLAMP 4.135166556742356f    // log(1000/16)

typedef __attribute__((ext_vector_type(2))) float v2f;
typedef __attribute__((ext_vector_type(8))) float v8f;

__device__ __forceinline__ unsigned fkey(float f) {
  unsigned u = __float_as_uint(f);
  return (u & 0x80000000u) ? ~u : (u | 0x80000000u);
}

// ---------------- zero / init scratch (runs every launch: graph-safe) -------
__global__ void k_zero(unsigned* __restrict__ hist, unsigned* __restrict__ cnts,
                       float* __restrict__ candS, unsigned* __restrict__ candI) {
  int i = blockIdx.x * 256 + threadIdx.x;           // 160*256 = 40960 threads
  const int nh = NIMG * NLVL * NBINS;               // 40960
  if (i < nh) hist[i] = 0u;
  if (i < NIMG * NLVL * 2) cnts[i] = 0u;
  if (i < NIMG * NCAND) { candS[i] = -INFINITY; candI[i] = 0u; }
}

// ---------------- pass 1: per (img,level) histogram of logit keys -----------
// float4 (global_load_b128) streaming: 512B per wave per VMEM op to feed the
// 23.3 TB/s HBM pipe; LDS-privatized histogram, one global flush per block.
__global__ void k_hist(const float* __restrict__ lg, int perImg,
                       unsigned* __restrict__ histL) {
  __shared__ unsigned lh[NBINS];
  for (int b = threadIdx.x; b < NBINS; b += blockDim.x) lh[b] = 0u;
  __syncthreads();
  const int img = blockIdx.y;
  const float4* src4 = (const float4*)(lg + (size_t)img * perImg);
  unsigned* gh = histL + (size_t)img * (NLVL * NBINS);
  const int nvec = perImg >> 2;                     // perImg divisible by 4
  const int stride = gridDim.x * blockDim.x;
  for (int p = blockIdx.x * blockDim.x + threadIdx.x; p < nvec; p += stride) {
    __builtin_prefetch(src4 + p + stride, 0, 1);    // global_prefetch_b8
    float4 v = src4[p];
    if (v.x > LOGIT_TH) atomicAdd(&lh[fkey(v.x) >> 20], 1u);
    if (v.y > LOGIT_TH) atomicAdd(&lh[fkey(v.y) >> 20], 1u);
    if (v.z > LOGIT_TH) atomicAdd(&lh[fkey(v.z) >> 20], 1u);
    if (v.w > LOGIT_TH) atomicAdd(&lh[fkey(v.w) >> 20], 1u);
  }
  __syncthreads();
  for (int b = threadIdx.x; b < NBINS; b += blockDim.x)
    if (lh[b]) atomicAdd(&gh[b], lh[b]);
}

// ---------------- threshold selection: WMMA suffix-scan over 4096 bins ------
// Chunk totals t16[j] (j=0..255, 16 bins each) arranged column-major into a
// 16x16 matrix T[k][c] = t16[c*16+k].  sfx = U x T (U upper-triangular ones)
// gives the within-column suffix sums; chained 4x v_wmma_f32_16x16x4_f32
// (K = 16) keeps counts exact in f32 (max count ~11.8M < 2^24).
__global__ void k_thresh(const unsigned* __restrict__ hist, int* __restrict__ thr) {
  const int il = blockIdx.x;                        // img*NLVL + lvl
  const unsigned* h = hist + (size_t)il * NBINS;
  __shared__ float hf[NBINS];
  __shared__ float t16[256];
  __shared__ float sfx[256];
  __shared__ float ct[16];
  for (int b = threadIdx.x; b < NBINS; b += 256) hf[b] = (float)h[b];
  __syncthreads();
  {
    float s = 0.f;
    const int base = threadIdx.x * 16;
    #pragma unroll
    for (int k = 0; k < 16; ++k) s += hf[base + k];
    t16[threadIdx.x] = s;
  }
  __syncthreads();
  if (threadIdx.x < 32) {                           // full wave, EXEC all-ones
    const int lane = threadIdx.x;
    const int half = lane >> 4;                     // K-pair select per layout
    const int m = lane & 15;                        // row (A) / col (B)
    v8f acc = {};
    #pragma unroll
    for (int s = 0; s < 4; ++s) {
      const int k0 = 4 * s + (half ? 2 : 0);
      const int k1 = k0 + 1;
      v2f a, b;
      a.x = (k0 >= m) ? 1.0f : 0.0f;                // U[m][k] = (k >= m)
      a.y = (k1 >= m) ? 1.0f : 0.0f;
      b.x = t16[m * 16 + k0];                       // T[k][c], c == m (lane&15)
      b.y = t16[m * 16 + k1];
      acc = __builtin_amdgcn_wmma_f32_16x16x4_f32(
          false, a, false, b, (short)0, acc, false, false);
    }
    const int rbase = half * 8;
    #pragma unroll
    for (int v = 0; v < 8; ++v) sfx[m * 16 + rbase + v] = acc[v];
  }
  __syncthreads();
  if (threadIdx.x == 0) {
    float run = 0.f;
    for (int c = 15; c >= 0; --c) { ct[c] = run; run += sfx[c * 16 + 0]; }
    const float total = run;
    int bstar = 0;
    float cntAbove = 0.f;
    if (total < (float)TOPK) {
      bstar = 0; cntAbove = total - hf[0];          // keep everything
    } else {
      int jstar = 0;
      for (int j = 255; j >= 0; --j) {
        float F = sfx[j] + ct[j >> 4];              // suffix over chunks >= j
        if (F >= (float)TOPK) { jstar = j; break; }
      }
      const float base =
          (jstar == 255) ? 0.f : (sfx[jstar + 1] + ct[(jstar + 1) >> 4]);
      float acc = base;
      for (int b = jstar * 16 + 15; b >= jstar * 16; --b) {
        acc += hf[b];
        if (acc >= (float)TOPK) { bstar = b; cntAbove = acc - hf[b]; break; }
      }
    }
    int ca = (int)(cntAbove + 0.5f);
    int need = TOPK - ca; if (need < 0) need = 0;
    thr[il * 4 + 0] = bstar;
    thr[il * 4 + 1] = ca;
    thr[il * 4 + 2] = need;
  }
}

// ---------------- pass 2: compact the top-K candidates (L2-resident) --------
__global__ void k_compact(const float* __restrict__ lg, int perImg, int HW,
                          const int* __restrict__ thr, unsigned* __restrict__ cnts,
                          float* __restrict__ candS, unsigned* __restrict__ candI,
                          int lvl) {
  const int img = blockIdx.y;
  const int il = img * NLVL + lvl;
  const int T = thr[il * 4 + 0];
  const int above = thr[il * 4 + 1];
  const int need = thr[il * 4 + 2];
  const float4* src4 = (const float4*)(lg + (size_t)img * perImg);
  unsigned* c2 = cnts + il * 2;
  float* cs = candS + img * NCAND + lvl * TOPK;
  unsigned* ci = candI + img * NCAND + lvl * TOPK;
  const int nvec = perImg >> 2;
  const int stride = gridDim.x * blockDim.x;
  for (int pv = blockIdx.x * blockDim.x + threadIdx.x; pv < nvec; pv += stride) {
    __builtin_prefetch(src4 + pv + stride, 0, 1);
    float4 q = src4[pv];
    float e[4] = {q.x, q.y, q.z, q.w};
    #pragma unroll
    for (int j = 0; j < 4; ++j) {
      float v = e[j];
      if (v <= LOGIT_TH) continue;
      const int bin = (int)(fkey(v) >> 20);
      int slot = -1;
      if (bin > T) {
        unsigned s = atomicAdd(&c2[0], 1u);
        if ((int)s < above) slot = (int)s;          // exact by construction
      } else if (bin == T) {
        unsigned t = atomicAdd(&c2[1], 1u);
        if ((int)t < need) slot = above + (int)t;
      }
      if (slot >= 0 && slot < TOPK) {
        const int p = pv * 4 + j;
        const int chan = p / HW, pix = p % HW;      // (A*80, H, W) layout
        const int a = chan / NUM_CLS, cls = chan % NUM_CLS;
        ci[slot] = (unsigned)((pix * ANC + a) * NUM_CLS + cls);  // HWA*K index
        cs[slot] = 1.0f / (1.0f + __expf(-v));
      }
    }
  }
}

// ---------------- decode boxes, clip, add class offset ----------------------
__global__ void k_decode(const float* __restrict__ d3, const float* __restrict__ d4,
                         const float* __restrict__ d5, const float* __restrict__ d6,
                         const float* __restrict__ d7, const float* __restrict__ anchors,
                         const float* __restrict__ candS, const unsigned* __restrict__ candI,
                         float* __restrict__ boxes, float* __restrict__ nboxes,
                         float* __restrict__ nscore, int* __restrict__ carr) {
  const int g = blockIdx.x * blockDim.x + threadIdx.x;
  if (g >= NIMG * NCAND) return;
  const int img = g / NCAND, r = g % NCAND;
  const int lvl = r / TOPK;
  const int HWs[NLVL]  = {128, 64, 32, 16, 8};
  const int AOFF[NLVL] = {0, 147456, 184320, 193536, 195840};
  const float* dls[NLVL] = {d3, d4, d5, d6, d7};
  const int W = HWs[lvl], HW = W * W;
  const float s = candS[g];
  const unsigned idx = candI[g];
  float ox1 = 0.f, oy1 = 0.f, ox2 = 0.f, oy2 = 0.f;
  float nx1 = -1e9f, ny1 = -1e9f, nx2 = -1e9f, ny2 = -1e9f;
  int cls = 0;
  if (s != -INFINITY) {
    cls = (int)(idx % NUM_CLS);
    const unsigned hwa = idx / NUM_CLS;
    const int a = (int)(hwa % ANC);
    const int pix = (int)(hwa / ANC);
    const float* dl = dls[lvl] + (size_t)img * (ANC * 4 * HW);
    float dx = dl[(a * 4 + 0) * HW + pix];
    float dy = dl[(a * 4 + 1) * HW + pix];
    float dw = fminf(dl[(a * 4 + 2) * HW + pix], SCALE_CLAMP);
    float dh = fminf(dl[(a * 4 + 3) * HW + pix], SCALE_CLAMP);
    const float* anc = anchors + (size_t)(AOFF[lvl] + hwa) * 4;
    const float aw = anc[2] - anc[0], ah = anc[3] - anc[1];
    const float cx = anc[0] + 0.5f * aw, cy = anc[1] + 0.5f * ah;
    const float pcx = dx * aw + cx, pcy = dy * ah + cy;
    const float pw = __expf(dw) * aw, ph = __expf(dh) * ah;
    ox1 = fminf(fmaxf(pcx - 0.5f * pw, 0.f), IMGSZ);
    oy1 = fminf(fmaxf(pcy - 0.5f * ph, 0.f), IMGSZ);
    ox2 = fminf(fmaxf(pcx + 0.5f * pw, 0.f), IMGSZ);
    oy2 = fminf(fmaxf(pcy + 0.5f * ph, 0.f), IMGSZ);
    const float off = (float)cls * (IMGSZ + 1.0f);
    nx1 = ox1 + off; ny1 = oy1 + off; nx2 = ox2 + off; ny2 = oy2 + off;
  }
  float4 ob = {ox1, oy1, ox2, oy2};
  float4 nb = {nx1, ny1, nx2, ny2};
  ((float4*)boxes)[g] = ob;
  ((float4*)nboxes)[g] = nb;
  nscore[g] = s;
  carr[g] = cls;
}

// ---------------- greedy NMS: one 1024-thread WG per image, LDS-resident ----
// 100 KB of dynamic LDS (legal: CDNA5 WGP has 320 KB) holds all 5000
// class-offset boxes + scores; 100 iterations of argmax + IoU suppression.
__global__ void __launch_bounds__(1024)
k_nms(const float* __restrict__ nboxes, const float* __restrict__ nscore,
      int* __restrict__ keep, float* __restrict__ kept) {
  extern __shared__ unsigned char smem[];
  float4* sb = (float4*)smem;                       // NCAND boxes   (80 KB)
  float* ss = (float*)(sb + NCAND);                 // NCAND scores  (20 KB)
  __shared__ float rVal[1024];
  __shared__ int rIdx[1024];
  __shared__ float4 bb;
  const int img = blockIdx.x, tid = threadIdx.x;
  for (int i = tid; i < NCAND; i += 1024) {
    sb[i] = ((const float4*)nboxes)[img * NCAND + i];
    ss[i] = nscore[img * NCAND + i];
  }
  __syncthreads();
  for (int it = 0; it < MAXDET; ++it) {
    float best = -INFINITY; int bidx = NCAND;
    for (int i = tid; i < NCAND; i += 1024) {
      float v = ss[i];
      if (v > best || (v == best && i < bidx)) { best = v; bidx = i; }
    }
    rVal[tid] = best; rIdx[tid] = bidx;
    __syncthreads();
    for (int w = 512; w > 0; w >>= 1) {             // first-max tie-break
      if (tid < w) {
        float v2 = rVal[tid + w]; int i2 = rIdx[tid + w];
        if (v2 > rVal[tid] || (v2 == rVal[tid] && i2 < rIdx[tid])) {
          rVal[tid] = v2; rIdx[tid] = i2;
        }
      }
      __syncthreads();
    }
    if (tid == 0) {
      int bi = rIdx[0];
      bb = sb[bi];
      keep[img * MAXDET + it] = bi;
      kept[img * MAXDET + it] = rVal[0];
      ss[bi] = -INFINITY;
    }
    __syncthreads();
    const float4 B = bb;
    const float areaB = (B.z - B.x) * (B.w - B.y);
    for (int i = tid; i < NCAND; i += 1024) {
      float4 q = sb[i];
      float inter = fmaxf(fminf(B.z, q.z) - fmaxf(B.x, q.x), 0.f) *
                    fmaxf(fminf(B.w, q.w) - fmaxf(B.y, q.y), 0.f);
      float uni = areaB + (q.z - q.x) * (q.w - q.y) - inter;
      float iou = (uni > 0.f) ? (inter / fmaxf(uni, 1e-9f)) : 0.f;
      if (iou > NMS_IOU) ss[i] = -INFINITY;
    }
    __syncthreads();
  }
}

// ---------------- gather outputs: boxes(2,100,4) | scores(2,100) | cls(2,100)
__global__ void k_final(const float* __restrict__ boxes, const int* __restrict__ carr,
                        const int* __restrict__ keep, const float* __restrict__ kept,
                        float* __restrict__ out) {
  const int g = blockIdx.x * blockDim.x + threadIdx.x;
  if (g >= NIMG * MAXDET) return;
  const int img = g / MAXDET;
  const int ki = keep[g];
  const float kv = kept[g];
  const bool valid = (kv != -INFINITY);
  const float* b = boxes + (size_t)(img * NCAND + ki) * 4;
  float* bo = out + (size_t)g * 4;
  bo[0] = valid ? b[0] : 0.f;
  bo[1] = valid ? b[1] : 0.f;
  bo[2] = valid ? b[2] : 0.f;
  bo[3] = valid ? b[3] : 0.f;
  out[NIMG * MAXDET * 4 + g] = valid ? kv : 0.f;
  out[NIMG * MAXDET * 5 + g] = valid ? (float)carr[img * NCAND + ki] : -1.0f;
}

extern "C" void kernel_launch(void* const* d_in, const int* in_sizes, int n_in,
                              void* d_out, int out_size, void* d_ws, size_t ws_size,
                              hipStream_t stream) {
  (void)in_sizes; (void)n_in; (void)out_size; (void)ws_size;
  // setup_inputs() dict order: logits_p3, deltas_p3, logits_p4, deltas_p4, ...
  const float* lg[NLVL]; const float* dl[NLVL];
  for (int i = 0; i < NLVL; ++i) {
    lg[i] = (const float*)d_in[2 * i];
    dl[i] = (const float*)d_in[2 * i + 1];
  }
  const float* anchors = (const float*)d_in[10];
  float* outF = (float*)d_out;

  unsigned char* ws = (unsigned char*)d_ws;
  size_t off = 0;
  auto alloc = [&](size_t bytes) -> void* {
    void* p = ws + off; off += (bytes + 255) & ~(size_t)255; return p;
  };
  unsigned* hist  = (unsigned*)alloc((size_t)NIMG * NLVL * NBINS * 4);
  int*      thr   = (int*)     alloc((size_t)NIMG * NLVL * 4 * 4);
  unsigned* cnts  = (unsigned*)alloc((size_t)NIMG * NLVL * 2 * 4);
  float*    candS = (float*)   alloc((size_t)NIMG * NCAND * 4);
  unsigned* candI = (unsigned*)alloc((size_t)NIMG * NCAND * 4);
  float*    boxes = (float*)   alloc((size_t)NIMG * NCAND * 4 * 4);
  float*    nbox  = (float*)   alloc((size_t)NIMG * NCAND * 4 * 4);
  float*    nsc   = (float*)   alloc((size_t)NIMG * NCAND * 4);
  int*      carr  = (int*)     alloc((size_t)NIMG * NCAND * 4);
  int*      keep  = (int*)     alloc((size_t)NIMG * MAXDET * 4);
  float*    kept  = (float*)   alloc((size_t)NIMG * MAXDET * 4);

  static const int HWs[NLVL] = {128, 64, 32, 16, 8};

  k_zero<<<dim3(160), dim3(256), 0, stream>>>(hist, cnts, candS, candI);

  for (int l = 0; l < NLVL; ++l) {
    const int HW = HWs[l] * HWs[l];
    const int perImg = ANC * NUM_CLS * HW;
    const int nvec = perImg / 4;
    int blocks = (nvec + 256 * 8 - 1) / (256 * 8);
    if (blocks > 1024) blocks = 1024;
    k_hist<<<dim3(blocks, NIMG), dim3(256), 0, stream>>>(lg[l], perImg,
                                                         hist + l * NBINS);
  }

  k_thresh<<<dim3(NIMG * NLVL), dim3(256), 0, stream>>>(hist, thr);

  for (int l = 0; l < NLVL; ++l) {
    const int HW = HWs[l] * HWs[l];
    const int perImg = ANC * NUM_CLS * HW;
    const int nvec = perImg / 4;
    int blocks = (nvec + 256 * 8 - 1) / (256 * 8);
    if (blocks > 1024) blocks = 1024;
    k_compact<<<dim3(blocks, NIMG), dim3(256), 0, stream>>>(
        lg[l], perImg, HW, thr, cnts, candS, candI, l);
  }

  k_decode<<<(NIMG * NCAND + 255) / 256, 256, 0, stream>>>(
      dl[0], dl[1], dl[2], dl[3], dl[4], anchors, candS, candI,
      boxes, nbox, nsc, carr);

  const size_t nmsShm = (size_t)NCAND * sizeof(float4) + (size_t)NCAND * sizeof(float);
  k_nms<<<dim3(NIMG), dim3(1024), nmsShm, stream>>>(nbox, nsc, keep, kept);

  k_final<<<(NIMG * MAXDET + 255) / 256, 256, 0, stream>>>(boxes, carr, keep,
                                                           kept, outF);
}